// ConvGNN_22454089023566
// MI455X (gfx1250) — compile-verified
//
#include <hip/hip_runtime.h>

typedef __attribute__((ext_vector_type(2))) float v2f;
typedef __attribute__((ext_vector_type(8))) float v8f;

#define N_NODES   50000
#define N_EDGES   600000
#define F         128
#define N_GRAPHS  64
#define N_CLASSES 10

// ---------------------------------------------------------------------------
// zero fill
// ---------------------------------------------------------------------------
__global__ void zero_f32(float* __restrict__ p, int n) {
    int i = blockIdx.x * blockDim.x + threadIdx.x;
    int stride = gridDim.x * blockDim.x;
    for (; i < n; i += stride) p[i] = 0.0f;
}

// ---------------------------------------------------------------------------
// edge scatter: agg[dst] += hin[src]   (each thread: one edge x 4 features)
// ---------------------------------------------------------------------------
__global__ __launch_bounds__(256) void edge_scatter(
    const float* __restrict__ hin, const int* __restrict__ src,
    const int* __restrict__ dst, float* __restrict__ agg)
{
    long long gid = (long long)blockIdx.x * blockDim.x + threadIdx.x;
    int e  = (int)(gid >> 5);          // 32 threads per edge (128 feats / 4)
    int f4 = ((int)gid & 31) << 2;
    if (e >= N_EDGES) return;
    int s = src[e], d = dst[e];
    if ((unsigned)s >= N_NODES || (unsigned)d >= N_NODES) return;  // safety
    const float4 v = *(const float4*)(hin + (long long)s * F + f4);
    float* o = agg + (long long)d * F + f4;
    atomicAdd(o + 0, v.x);
    atomicAdd(o + 1, v.y);
    atomicAdd(o + 2, v.z);
    atomicAdd(o + 3, v.w);
}

// ---------------------------------------------------------------------------
// fused GraphConv linear:  out = [relu]( agg @ Wrel^T + bias + xin @ Wroot^T )
// One wave = one 16x16 output tile, V_WMMA_F32_16X16X4_F32, K loop of 32.
// Block = 256 threads = 8 waves = 8 N-tiles of a 16-row stripe.
// ---------------------------------------------------------------------------
__global__ __launch_bounds__(256) void gconv_gemm(
    const float* __restrict__ agg,  const float* __restrict__ xin,
    const float* __restrict__ Wrel, const float* __restrict__ Wroot,
    const float* __restrict__ bias, float* __restrict__ out, int relu)
{
    const int lane = threadIdx.x & 31;
    const int wave = threadIdx.x >> 5;      // 0..7 -> N tile index
    const int m0   = blockIdx.x * 16;       // 3125 blocks * 16 = 50000
    const int n0   = wave * 16;
    const int half = lane >> 4;             // 0: K=k..k+1, 1: K=k+2..k+3
    const int l16  = lane & 15;

    // A (16x4 fp32): lanes 0-15 rows m0+l16 cols k..k+1; lanes 16-31 cols k+2..k+3
    const float* aP = agg  + (long long)(m0 + l16) * F + 2 * half;
    const float* xP = xin  + (long long)(m0 + l16) * F + 2 * half;
    // B (4x16): B[k][n] = W[n][k]; same float2 addressing pattern on W rows
    const float* bRel  = Wrel  + (long long)(n0 + l16) * F + 2 * half;
    const float* bRoot = Wroot + (long long)(n0 + l16) * F + 2 * half;

    const float bval = bias[n0 + l16];
    v8f c0, c1;
#pragma unroll
    for (int r = 0; r < 8; ++r) { c0[r] = bval; c1[r] = 0.0f; }

#pragma unroll 8
    for (int k = 0; k < F; k += 4) {
        v2f a  = *(const v2f*)(aP + k);
        v2f b  = *(const v2f*)(bRel + k);
        c0 = __builtin_amdgcn_wmma_f32_16x16x4_f32(
                false, a, false, b, (short)0, c0, false, false);
        v2f a2 = *(const v2f*)(xP + k);
        v2f b2 = *(const v2f*)(bRoot + k);
        c1 = __builtin_amdgcn_wmma_f32_16x16x4_f32(
                false, a2, false, b2, (short)0, c1, false, false);
    }
    v8f c = c0 + c1;

    // C/D layout: VGPR r -> row (r + 8*half), column n0 + l16
    float* oP = out + (long long)(m0 + 8 * half) * F + n0 + l16;
#pragma unroll
    for (int r = 0; r < 8; ++r) {
        float v = c[r];
        if (relu) v = fmaxf(v, 0.0f);
        oP[(long long)r * F] = v;
    }
}

// ---------------------------------------------------------------------------
// segment mean pool part 1: atomic sums + counts
// ---------------------------------------------------------------------------
__global__ __launch_bounds__(256) void pool_scatter(
    const float* __restrict__ h, const int* __restrict__ batch,
    float* __restrict__ psum, float* __restrict__ pcnt)
{
    long long gid = (long long)blockIdx.x * blockDim.x + threadIdx.x;
    int n  = (int)(gid >> 5);
    int f4 = ((int)gid & 31) << 2;
    if (n >= N_NODES) return;
    int g = batch[n];
    if ((unsigned)g >= N_GRAPHS) return;   // safety
    const float4 v = *(const float4*)(h + (long long)n * F + f4);
    float* o = psum + (long long)g * F + f4;
    atomicAdd(o + 0, v.x);
    atomicAdd(o + 1, v.y);
    atomicAdd(o + 2, v.z);
    atomicAdd(o + 3, v.w);
    if (f4 == 0) atomicAdd(&pcnt[g], 1.0f);
}

// ---------------------------------------------------------------------------
// final: out[g][c] = b_lin[c] + sum_f (psum[g][f] / max(cnt[g],1)) * W_lin[c][f]
// ---------------------------------------------------------------------------
__global__ void final_linear(
    const float* __restrict__ psum, const float* __restrict__ pcnt,
    const float* __restrict__ Wlin, const float* __restrict__ blin,
    float* __restrict__ out)
{
    int t = threadIdx.x;                 // 640 threads
    if (t >= N_GRAPHS * N_CLASSES) return;
    int g = t / N_CLASSES;
    int c = t % N_CLASSES;
    float inv = 1.0f / fmaxf(pcnt[g], 1.0f);
    float acc = blin[c];
    const float* pg = psum + g * F;
    const float* wc = Wlin + c * F;
#pragma unroll 4
    for (int f = 0; f < F; ++f) acc += pg[f] * inv * wc[f];
    out[g * N_CLASSES + c] = acc;
}

// ---------------------------------------------------------------------------
extern "C" void kernel_launch(void* const* d_in, const int* in_sizes, int n_in,
                              void* d_out, int out_size, void* d_ws, size_t ws_size,
                              hipStream_t stream) {
    const float* x     = (const float*)d_in[0];
    const int*   edge  = (const int*)d_in[1];
    const int*   src   = edge;
    const int*   dst   = edge + N_EDGES;
    const int*   batch = (const int*)d_in[2];
    const float* W1r = (const float*)d_in[3];
    const float* b1  = (const float*)d_in[4];
    const float* W1o = (const float*)d_in[5];
    const float* W2r = (const float*)d_in[6];
    const float* b2  = (const float*)d_in[7];
    const float* W2o = (const float*)d_in[8];
    const float* W3r = (const float*)d_in[9];
    const float* b3  = (const float*)d_in[10];
    const float* W3o = (const float*)d_in[11];
    const float* Wlin = (const float*)d_in[12];
    const float* blin = (const float*)d_in[13];
    float* out = (float*)d_out;

    // workspace layout
    float* agg  = (float*)d_ws;                 // N_NODES*F
    float* hA   = agg + (long long)N_NODES * F; // N_NODES*F
    float* hB   = hA  + (long long)N_NODES * F; // N_NODES*F
    float* psum = hB  + (long long)N_NODES * F; // N_GRAPHS*F
    float* pcnt = psum + N_GRAPHS * F;          // N_GRAPHS (contiguous with psum)

    const int scatterBlocks = (N_EDGES * 32 + 255) / 256;   // 75000
    const int gemmBlocks    = N_NODES / 16;                 // 3125
    const int poolBlocks    = (N_NODES * 32 + 255) / 256;

    // ---- layer 1 ----
    zero_f32<<<2048, 256, 0, stream>>>(agg, N_NODES * F);
    edge_scatter<<<scatterBlocks, 256, 0, stream>>>(x, src, dst, agg);
    gconv_gemm<<<gemmBlocks, 256, 0, stream>>>(agg, x, W1r, W1o, b1, hA, 1);
    // ---- layer 2 ----
    zero_f32<<<2048, 256, 0, stream>>>(agg, N_NODES * F);
    edge_scatter<<<scatterBlocks, 256, 0, stream>>>(hA, src, dst, agg);
    gconv_gemm<<<gemmBlocks, 256, 0, stream>>>(agg, hA, W2r, W2o, b2, hB, 1);
    // ---- layer 3 (no relu) ----
    zero_f32<<<2048, 256, 0, stream>>>(agg, N_NODES * F);
    edge_scatter<<<scatterBlocks, 256, 0, stream>>>(hB, src, dst, agg);
    gconv_gemm<<<gemmBlocks, 256, 0, stream>>>(agg, hB, W3r, W3o, b3, hA, 0);
    // ---- mean pool + classifier ----
    zero_f32<<<64, 256, 0, stream>>>(psum, N_GRAPHS * F + N_GRAPHS);
    pool_scatter<<<poolBlocks, 256, 0, stream>>>(hA, batch, psum, pcnt);
    final_linear<<<1, 640, 0, stream>>>(psum, pcnt, Wlin, blin, out);
}